// EigenEstimation_40802189312446
// MI455X (gfx1250) — compile-verified
//
#include <hip/hip_runtime.h>

// Problem constants (match reference)
#define B_    256
#define DIN_  1024
#define DH_   4096
#define DOUT_ 1024
#define K_    16

typedef __attribute__((ext_vector_type(2))) float v2f;
typedef __attribute__((ext_vector_type(4))) float f4;
typedef __attribute__((ext_vector_type(8))) float v8f;

// D = A(16x4,f32) * B(4x16,f32) + C(16x16,f32)  — CDNA5 fp32 WMMA
static __device__ __forceinline__ v8f wmma4(v2f a, v2f b, v8f c) {
  return __builtin_amdgcn_wmma_f32_16x16x4_f32(false, a, false, b, (short)0, c, false, false);
}

// ---------------------------------------------------------------------------
// Stage 1: h = relu(x @ W1^T + b1)   (256 x 4096)
// grid(8,16) x 256thr: block.y = m-tile, block.x*8waves -> 64-wide n strips
// ---------------------------------------------------------------------------
__global__ __launch_bounds__(256) void k_h(const float* __restrict__ x,
                                           const float* __restrict__ W1,
                                           const float* __restrict__ b1,
                                           float* __restrict__ h) {
  const int wave = threadIdx.x >> 5, lane = threadIdx.x & 31;
  const int half = lane >> 4, r16 = lane & 15;
  const int m0 = blockIdx.y << 4;
  const int n0 = (blockIdx.x << 9) + (wave << 6);

  const float* ap = x + (size_t)(m0 + r16) * DIN_ + (half << 1);
  const float* bp[4];
#pragma unroll
  for (int t = 0; t < 4; ++t)
    bp[t] = W1 + (size_t)(n0 + (t << 4) + r16) * DIN_ + (half << 1);

  v8f acc[4] = {};
  for (int k = 0; k < DIN_; k += 4) {
    v2f a = *(const v2f*)(ap + k);
#pragma unroll
    for (int t = 0; t < 4; ++t) {
      v2f b = *(const v2f*)(bp[t] + k);
      acc[t] = wmma4(a, b, acc[t]);
    }
  }
#pragma unroll
  for (int t = 0; t < 4; ++t) {
    const int n = n0 + (t << 4) + r16;
    const float bias = b1[n];
#pragma unroll
    for (int r = 0; r < 8; ++r) {
      float v = acc[t][r] + bias;
      v = v > 0.f ? v : 0.f;
      h[(size_t)(m0 + (half << 3) + r) * DH_ + n] = v;
    }
  }
}

// ---------------------------------------------------------------------------
// Stage 2: dh[k] = (h>0) * (x @ uW1_k^T + ub1_k)   for all k (grid.z)
// ---------------------------------------------------------------------------
__global__ __launch_bounds__(256) void k_dh(const float* __restrict__ x,
                                            const float* __restrict__ uW1,
                                            const float* __restrict__ ub1,
                                            const float* __restrict__ h,
                                            float* __restrict__ dh) {
  const int wave = threadIdx.x >> 5, lane = threadIdx.x & 31;
  const int half = lane >> 4, r16 = lane & 15;
  const int kk = blockIdx.z;
  const int m0 = blockIdx.y << 4;
  const int n0 = (blockIdx.x << 9) + (wave << 6);

  const float* ap = x + (size_t)(m0 + r16) * DIN_ + (half << 1);
  const float* bp[4];
#pragma unroll
  for (int t = 0; t < 4; ++t)
    bp[t] = uW1 + ((size_t)kk * DH_ + n0 + (t << 4) + r16) * DIN_ + (half << 1);

  v8f acc[4] = {};
  for (int k = 0; k < DIN_; k += 4) {
    v2f a = *(const v2f*)(ap + k);
#pragma unroll
    for (int t = 0; t < 4; ++t) {
      v2f b = *(const v2f*)(bp[t] + k);
      acc[t] = wmma4(a, b, acc[t]);
    }
  }
#pragma unroll
  for (int t = 0; t < 4; ++t) {
    const int n = n0 + (t << 4) + r16;
    const float bias = ub1[kk * DH_ + n];
#pragma unroll
    for (int r = 0; r < 8; ++r) {
      const int m = m0 + (half << 3) + r;
      float v = acc[t][r] + bias;
      const float hv = h[(size_t)m * DH_ + n];
      dh[((size_t)kk * B_ + m) * DH_ + n] = hv > 0.f ? v : 0.f;
    }
  }
}

// ---------------------------------------------------------------------------
// Stage 3: dout = dh_k @ W2^T + h @ uW2_k^T + ub2_k ; partial row sum-of-squares
// grid(2,16,16): x*8waves -> 16 n-tiles of 64, y = m-tile, z = k
// partial[((k*16+mt)*16+nt)*16 + localRow]
// ---------------------------------------------------------------------------
__global__ __launch_bounds__(256) void k_dout(const float* __restrict__ dh,
                                              const float* __restrict__ h,
                                              const float* __restrict__ W2,
                                              const float* __restrict__ uW2,
                                              const float* __restrict__ ub2,
                                              float* __restrict__ partial) {
  const int wave = threadIdx.x >> 5, lane = threadIdx.x & 31;
  const int half = lane >> 4, r16 = lane & 15;
  const int kk = blockIdx.z;
  const int m0 = blockIdx.y << 4;
  const int ntile = (blockIdx.x << 3) + wave;
  const int n0 = ntile << 6;

  const float* a1 = dh + ((size_t)kk * B_ + m0 + r16) * DH_ + (half << 1);
  const float* a2 = h + (size_t)(m0 + r16) * DH_ + (half << 1);
  const float* b1p[4];
  const float* b2p[4];
#pragma unroll
  for (int t = 0; t < 4; ++t) {
    b1p[t] = W2 + (size_t)(n0 + (t << 4) + r16) * DH_ + (half << 1);
    b2p[t] = uW2 + ((size_t)kk * DOUT_ + n0 + (t << 4) + r16) * DH_ + (half << 1);
  }

  v8f acc[4] = {};
  for (int k = 0; k < DH_; k += 4) {
    v2f x1 = *(const v2f*)(a1 + k);
    v2f x2 = *(const v2f*)(a2 + k);
#pragma unroll
    for (int t = 0; t < 4; ++t) {
      v2f bb1 = *(const v2f*)(b1p[t] + k);
      v2f bb2 = *(const v2f*)(b2p[t] + k);
      acc[t] = wmma4(x1, bb1, acc[t]);
      acc[t] = wmma4(x2, bb2, acc[t]);
    }
  }

  float p[8] = {0.f, 0.f, 0.f, 0.f, 0.f, 0.f, 0.f, 0.f};
#pragma unroll
  for (int t = 0; t < 4; ++t) {
    const float bias = ub2[kk * DOUT_ + n0 + (t << 4) + r16];
#pragma unroll
    for (int r = 0; r < 8; ++r) {
      const float v = acc[t][r] + bias;
      p[r] += v * v;
    }
  }
  // reduce over the 16 lanes of each half-wave (each half owns 8 rows)
#pragma unroll
  for (int r = 0; r < 8; ++r) {
    float s = p[r];
    s += __shfl_xor(s, 1, 16);
    s += __shfl_xor(s, 2, 16);
    s += __shfl_xor(s, 4, 16);
    s += __shfl_xor(s, 8, 16);
    p[r] = s;
  }
  if (r16 == 0) {
    const int base = (((kk << 4) + blockIdx.y) * 16 + ntile) * 16 + (half << 3);
#pragma unroll
    for (int r = 0; r < 8; ++r) partial[base + r] = p[r];
  }
}

// ---------------------------------------------------------------------------
// Stage 4: dH[m*K + k] = 2 * sum_nt partial[...]   (deterministic reduce)
// ---------------------------------------------------------------------------
__global__ __launch_bounds__(256) void k_reduce(const float* __restrict__ partial,
                                                float* __restrict__ dH) {
  const int tid = blockIdx.x * 256 + threadIdx.x; // 4096 total
  const int kk = tid >> 8;
  const int m = tid & 255;
  const int mt = m >> 4, r = m & 15;
  float s = 0.f;
#pragma unroll
  for (int nt = 0; nt < 16; ++nt)
    s += partial[(((kk << 4) + mt) * 16 + nt) * 16 + r];
  dH[m * K_ + kk] = 2.0f * s;
}

// ---------------------------------------------------------------------------
// u_tensor concatenation copy (nontemporal float4 streaming)
// dst[k*P4 + off4 + i] = src[k*seg4 + i]
// ---------------------------------------------------------------------------
__global__ __launch_bounds__(256) void k_copy(const f4* __restrict__ src,
                                              f4* __restrict__ dst,
                                              long long seg4, long long P4,
                                              long long off4) {
  const long long i = (long long)blockIdx.x * 256 + threadIdx.x;
  if (i >= seg4) return;
  const long long k = blockIdx.y;
  f4 v = __builtin_nontemporal_load(&src[k * seg4 + i]);
  __builtin_nontemporal_store(v, &dst[k * P4 + off4 + i]);
}

extern "C" void kernel_launch(void* const* d_in, const int* in_sizes, int n_in,
                              void* d_out, int out_size, void* d_ws, size_t ws_size,
                              hipStream_t stream) {
  (void)in_sizes; (void)n_in; (void)d_ws; (void)ws_size; (void)out_size;
  const float* x   = (const float*)d_in[0];
  const float* W1  = (const float*)d_in[1];
  const float* b1  = (const float*)d_in[2];
  const float* W2  = (const float*)d_in[3];
  const float* b2  = (const float*)d_in[4]; (void)b2; // unused by the math
  const float* uW1 = (const float*)d_in[5];
  const float* ub1 = (const float*)d_in[6];
  const float* uW2 = (const float*)d_in[7];
  const float* ub2 = (const float*)d_in[8];

  float* out   = (float*)d_out;
  float* dH    = out;                 // (B, K) = 4096 floats
  float* ubase = out + (size_t)B_ * K_;   // u_tensor (K, P) region, 537 MB

  // Scratch lives in the u_tensor region; the final copy kernels (launched
  // last, stream-ordered) overwrite it with the correct u_tensor contents.
  float* s_h  = ubase;                                   // 256*4096   = 4 MB
  float* s_dh = s_h + (size_t)B_ * DH_;                  // 16*256*4096 = 64 MB
  float* s_p  = s_dh + (size_t)K_ * B_ * DH_;            // 16*16*16*16 = 256 KB

  const dim3 blk(256);
  k_h    <<<dim3(8, 16),     blk, 0, stream>>>(x, W1, b1, s_h);
  k_dh   <<<dim3(8, 16, 16), blk, 0, stream>>>(x, uW1, ub1, s_h, s_dh);
  k_dout <<<dim3(2, 16, 16), blk, 0, stream>>>(s_dh, s_h, W2, uW2, ub2, s_p);
  k_reduce<<<dim3(16),       blk, 0, stream>>>(s_p, dH);

  // u_tensor = concat([uW1, ub1, uW2, ub2] flattened per k), row stride P
  const long long sW1 = (long long)DH_ * DIN_ / 4;   // 1,048,576 float4
  const long long sb1 = DH_ / 4;                     // 1,024
  const long long sW2 = (long long)DOUT_ * DH_ / 4;  // 1,048,576
  const long long sb2 = DOUT_ / 4;                   // 256
  const long long P4  = sW1 + sb1 + sW2 + sb2;       // 2,098,432
  f4* dst = (f4*)ubase;

  k_copy<<<dim3((unsigned)((sW1 + 255) / 256), K_), blk, 0, stream>>>((const f4*)uW1, dst, sW1, P4, 0);
  k_copy<<<dim3((unsigned)((sb1 + 255) / 256), K_), blk, 0, stream>>>((const f4*)ub1, dst, sb1, P4, sW1);
  k_copy<<<dim3((unsigned)((sW2 + 255) / 256), K_), blk, 0, stream>>>((const f4*)uW2, dst, sW2, P4, sW1 + sb1);
  k_copy<<<dim3((unsigned)((sb2 + 255) / 256), K_), blk, 0, stream>>>((const f4*)ub2, dst, sb2, P4, sW1 + sb1 + sW2);
}